// ClassificationHead_33947421508158
// MI455X (gfx1250) — compile-verified
//
#include <hip/hip_runtime.h>

typedef __attribute__((ext_vector_type(2))) float v2f;
typedef __attribute__((ext_vector_type(8))) float v8f;

#define NTASKS 512
#define NQ     75
#define NS     25
#define DIM    1024
#define NWAY   5
#define NSHOT  5

#define KC     64          // K-chunk staged in LDS per wave
#define PSTR   1032        // padded proto row stride (floats), 1032 % 64 = 8 banks skew
#define QSTR   68          // padded qtile row stride (floats)
#define NWAVES 5
#define BLOCK  (NWAVES * 32)

__global__ __launch_bounds__(BLOCK)
void proto_head_wmma_kernel(const float* __restrict__ query,
                            const float* __restrict__ support,
                            const float* __restrict__ scale,
                            float* __restrict__ out)
{
    __shared__ float protos[NWAY * PSTR];          // ~20.6 KB
    __shared__ float qtile[NWAVES][16 * QSTR];     // ~21.8 KB (wave-private slices)
    __shared__ float bb[NWAY];                     // |proto|^2
    __shared__ float aa_lds[NWAVES * 16];          // |query|^2 per tile row

    const int b    = blockIdx.x;
    const int tid  = threadIdx.x;
    const int lane = tid & 31;
    const int wave = tid >> 5;

    const float* sup = support + (size_t)b * NS * DIM;
    const float* q   = query   + (size_t)b * NQ * DIM;

    if (tid < NWAY) bb[tid] = 0.0f;
    __syncthreads();

    // ---- Phase 1: prototypes (class means) + |p|^2, coalesced over columns ----
    float bbp[NWAY] = {0.f, 0.f, 0.f, 0.f, 0.f};
    for (int d = tid; d < DIM; d += BLOCK) {
#pragma unroll
        for (int w = 0; w < NWAY; ++w) {
            float s = 0.0f;
#pragma unroll
            for (int j = 0; j < NSHOT; ++j)
                s += sup[(size_t)(w * NSHOT + j) * DIM + d];
            float p = s * 0.2f;           // / n_shot
            protos[w * PSTR + d] = p;
            bbp[w] += p * p;
        }
    }
#pragma unroll
    for (int w = 0; w < NWAY; ++w) atomicAdd(&bb[w], bbp[w]);
    __syncthreads();

    // ---- Phase 2: each wave owns one 16-row query tile ----
    const int mbase = wave * 16;
    const int mrow  = lane & 15;          // A-matrix M row (and B-matrix N col)
    const int hi    = lane >> 4;          // 0: K 0..1, 1: K 2..3 (f32 16x4 layout)
    const int klo   = 2 * hi;

    const int   n     = lane & 15;        // class column for B/D
    const int   nclmp = (n < NWAY) ? n : (NWAY - 1);
    const float nmask = (n < NWAY) ? 1.0f : 0.0f;   // branch-free masking of pad cols

    float* qt = &qtile[wave][0];
    v8f   acc = {};
    float aa_part = 0.0f;

    for (int kc = 0; kc < DIM; kc += KC) {
        // Stage 16 x KC query tile: lane loads one float2 per row (coalesced 256B/row)
#pragma unroll 4
        for (int r = 0; r < 16; ++r) {
            int m = mbase + r;
            float2 v = make_float2(0.0f, 0.0f);
            if (m < NQ)
                v = *(const float2*)(q + (size_t)m * DIM + kc + lane * 2);
            qt[r * QSTR + lane * 2]     = v.x;
            qt[r * QSTR + lane * 2 + 1] = v.y;
        }

        // Prefetch next chunk of this lane's row (global_prefetch_b8; speculative-safe)
        if (kc + KC < DIM) {
            int mp = mbase + mrow;
            if (mp >= NQ) mp = NQ - 1;
            __builtin_prefetch(q + (size_t)mp * DIM + kc + KC + lane * 2, 0, 0);
        }

        // |q|^2 partial: lane covers half a row of this chunk (LDS in-order per wave)
        {
            float s = 0.0f;
#pragma unroll
            for (int c = 0; c < 32; ++c) {
                float v = qt[mrow * QSTR + hi * 32 + c];
                s += v * v;
            }
            aa_part += s;
        }

        // FP32 WMMA: D(16x16) += A(16x4) * B(4x16), K marched 4 at a time
#pragma unroll
        for (int k = 0; k < KC; k += 4) {
            v2f a, bf;
            a.x  = qt[mrow * QSTR + k + klo];
            a.y  = qt[mrow * QSTR + k + klo + 1];
            bf.x = nmask * protos[nclmp * PSTR + kc + k + klo];
            bf.y = nmask * protos[nclmp * PSTR + kc + k + klo + 1];
            acc = __builtin_amdgcn_wmma_f32_16x16x4_f32(
                /*neg_a=*/false, a, /*neg_b=*/false, bf,
                /*c_mod=*/(short)0, acc, /*reuse_a=*/false, /*reuse_b=*/false);
        }
    }

    // Combine the two K-halves of |q|^2 (wave32 shuffle), park per-row in LDS
    float aa_full = aa_part + __shfl_xor(aa_part, 16, 32);
    if (lane < 16) aa_lds[wave * 16 + lane] = aa_full;

    // ---- Epilogue: logits = scale * (2*AB - AA - BB) / D ----
    const float scl   = scale[0];
    const float inv_d = 1.0f / (float)DIM;
    if (n < NWAY) {
#pragma unroll
        for (int j = 0; j < 8; ++j) {
            int m = mbase + j + 8 * hi;   // C/D layout: VGPR j, lane L -> M = j + 8*(L>>4)
            if (m < NQ) {
                float aav   = aa_lds[wave * 16 + j + 8 * hi];
                float logit = (2.0f * acc[j] - aav - bb[n]) * inv_d * scl;
                out[((size_t)b * NQ + m) * NWAY + n] = logit;
            }
        }
    }
}

extern "C" void kernel_launch(void* const* d_in, const int* in_sizes, int n_in,
                              void* d_out, int out_size, void* d_ws, size_t ws_size,
                              hipStream_t stream) {
    // setup_inputs order: query, support, support_labels, n_way, n_shot, scale
    const float* query   = (const float*)d_in[0];
    const float* support = (const float*)d_in[1];
    const float* scale   = (const float*)d_in[5];
    float* out = (float*)d_out;

    proto_head_wmma_kernel<<<NTASKS, BLOCK, 0, stream>>>(query, support, scale, out);
}